// FitStepOutput_55611236549180
// MI455X (gfx1250) — compile-verified
//
#include <hip/hip_runtime.h>
#include <hip/hip_bf16.h>

// ---------------------------------------------------------------------------
// FitStepOutput: per-row exclusive cumsum(1-2x) -> first argmin over 129
// scores -> midpoint of two clamped bin edges.  Pure streaming kernel:
// 512MB in / 4MB out @ 23.3 TB/s  =>  ~22us floor.  Strategy: async
// global->LDS tile DMA (CDNA5 ASYNCcnt path), double-buffered per wave,
// lane-per-row serial scan from bank-padded LDS.
// ---------------------------------------------------------------------------

#define NBINS 128

constexpr int WAVE            = 32;
constexpr int WAVES_PER_BLOCK = 8;
constexpr int BLOCK_THREADS   = WAVES_PER_BLOCK * WAVE;   // 256
constexpr int ROW_W           = NBINS + 4;                // 132 words: 16B-aligned rows, banks spread
constexpr int TILE_ROWS       = 32;                       // rows per wave-tile (1 row per lane)
constexpr int TILE_W          = TILE_ROWS * ROW_W;        // 4224 words per buffer

typedef float float4v __attribute__((ext_vector_type(4)));
typedef int   int4v   __attribute__((ext_vector_type(4)));

#if defined(__HIP_DEVICE_COMPILE__) && defined(__gfx1250__) && \
    __has_builtin(__builtin_amdgcn_global_load_async_to_lds_b128)
#define USE_ASYNC_LDS 1
#else
#define USE_ASYNC_LDS 0
#endif

#if USE_ASYNC_LDS
// builtin signature (from clang diagnostic): (int4 AS1* src, int4 AS3* dst, imm off, imm cpol)
typedef __attribute__((address_space(1))) int4v* gv4_ptr;
typedef __attribute__((address_space(3))) int4v* lv4_ptr;
// async loads complete in order: cnt<=32 means the previous tile's 32 copies landed
#define WAIT_ASYNC_LE32() asm volatile("s_wait_asynccnt 0x20" ::: "memory")
#define WAIT_ASYNC_0()    asm volatile("s_wait_asynccnt 0x0"  ::: "memory")
#else
#define WAIT_ASYNC_LE32() ((void)0)
#define WAIT_ASYNC_0()    ((void)0)
#endif

// Copy one 32x128 f32 tile (16KB) into LDS with 132-word row stride.
// One instruction per row: 32 lanes x b128 = 512B, fully coalesced.
__device__ __forceinline__ void prefetch_tile(const float* __restrict__ gsrc,
                                              float* lbuf, int lane) {
#pragma unroll
  for (int r = 0; r < TILE_ROWS; ++r) {
    const float* g = gsrc + r * NBINS + lane * 4;
    float*       l = lbuf + r * ROW_W + lane * 4;
#if USE_ASYNC_LDS
    __builtin_amdgcn_global_load_async_to_lds_b128((gv4_ptr)g, (lv4_ptr)l, 0, 0);
#else
    *(float4v*)l = *(const float4v*)g;
#endif
  }
}

__global__ __launch_bounds__(BLOCK_THREADS)
void FitStepOutput_kernel(const float* __restrict__ in,
                          const float* __restrict__ bins,
                          float* __restrict__ out,
                          int nTiles) {
  __shared__ float tiles[2][WAVES_PER_BLOCK][TILE_W];   // ~270 KB, double-buffered
  __shared__ float sbins[NBINS];

  const int tid  = threadIdx.x;
  const int lane = tid & (WAVE - 1);
  const int w    = tid >> 5;

  if (tid < NBINS) sbins[tid] = bins[tid];
  __syncthreads();

  const int gw     = blockIdx.x * WAVES_PER_BLOCK + w;
  const int stride = gridDim.x * WAVES_PER_BLOCK;

  int t = gw;
  if (t < nTiles)
    prefetch_tile(in + (size_t)t * TILE_ROWS * NBINS, tiles[0][w], lane);

  int k = 0;
  for (; t < nTiles; t += stride, ++k) {
    const int tn = t + stride;
    if (tn < nTiles) {
      // software pipeline: start next tile's DMA, then wait for current tile
      prefetch_tile(in + (size_t)tn * TILE_ROWS * NBINS, tiles[(k + 1) & 1][w], lane);
      WAIT_ASYNC_LE32();
    } else {
      WAIT_ASYNC_0();
    }

    // lane-per-row serial scan + first-occurrence argmin over 129 scores
    const float* p = &tiles[k & 1][w][lane * ROW_W];
    float s  = 0.0f;
    float mv = 0.0f;   // scores[0] == 0
    int   mi = 0;
#pragma unroll
    for (int g4 = 0; g4 < NBINS / 4; ++g4) {
      const float4v v = *(const float4v*)(p + 4 * g4);
      float d;
      d = fmaf(-2.0f, v.x, 1.0f); s += d; if (s < mv) { mv = s; mi = 4 * g4 + 1; }
      d = fmaf(-2.0f, v.y, 1.0f); s += d; if (s < mv) { mv = s; mi = 4 * g4 + 2; }
      d = fmaf(-2.0f, v.z, 1.0f); s += d; if (s < mv) { mv = s; mi = 4 * g4 + 3; }
      d = fmaf(-2.0f, v.w, 1.0f); s += d; if (s < mv) { mv = s; mi = 4 * g4 + 4; }
    }

    // bins_to_select = [bins[0], bins..., bins[-1]];  chosen = avg of idx, idx+1
    const int i0 = (mi > 0) ? (mi - 1) : 0;
    const int i1 = (mi < NBINS) ? mi : (NBINS - 1);
    out[(size_t)t * TILE_ROWS + lane] = 0.5f * (sbins[i0] + sbins[i1]);
  }
}

extern "C" void kernel_launch(void* const* d_in, const int* in_sizes, int n_in,
                              void* d_out, int out_size, void* d_ws, size_t ws_size,
                              hipStream_t stream) {
  const float* input = (const float*)d_in[0];
  const float* bins  = (const float*)d_in[1];
  float*       out   = (float*)d_out;

  const int B      = in_sizes[0] / NBINS;          // 1048576
  const int nTiles = B / TILE_ROWS;                // 32768 wave-tiles (B divisible by 32)

  // ~8 tiles per wave -> enough blocks to cover the machine, few enough to pipeline
  int wavesTarget = (nTiles + 7) / 8;
  int grid = (wavesTarget + WAVES_PER_BLOCK - 1) / WAVES_PER_BLOCK;
  if (grid < 1) grid = 1;

  FitStepOutput_kernel<<<grid, BLOCK_THREADS, 0, stream>>>(input, bins, out, nTiles);
}